// MPOLinear3_6528350290551
// MI455X (gfx1250) — compile-verified
//
#include <hip/hip_runtime.h>

typedef __attribute__((ext_vector_type(2))) float v2f;
typedef __attribute__((ext_vector_type(4))) float v4f;
typedef __attribute__((ext_vector_type(8))) float v8f;

typedef unsigned int u32x4 __attribute__((ext_vector_type(4)));
typedef int          i32x4 __attribute__((ext_vector_type(4)));
typedef int          i32x8 __attribute__((ext_vector_type(8)));

#define LEG   16
#define RANK  64
#define TB    8                 // samples per block (one per wave for TDM staging)
#define RPAD  18                // padded LDS row stride in floats (even -> 8B aligned pairs)
#define SSTRIDE (256 * RPAD)    // floats per sample per LDS buffer (4608)

// --- CDNA5 TDM availability / arity detection --------------------------------
#if defined(__has_builtin)
#if __has_builtin(__builtin_amdgcn_tensor_load_to_lds) && \
    __has_builtin(__builtin_amdgcn_s_wait_tensorcnt)
#define HAVE_TDM 1
#endif
#endif
#if defined(HAVE_TDM)
#if __has_include(<hip/amd_detail/amd_gfx1250_TDM.h>)
#define TDM_ARGS6 1   // therock-10.0 headers -> 6-arg builtin
#endif
#endif

// ---------------------------------------------------------------------------
// Kernel 1: collapse the rank dimensions of the three MPO cores.
//   L[c][z] = sum_j last[c,j,z]             -> ws[0..255]
//   M[b][y] = sum_{r1,r2} middle[b,r1,r2,y] -> ws[256..511]
//   F[a][x] = sum_j first[a,j,x]            -> ws[512..767]
// Layout [in][out] row-major == exactly the B-operand (K=in rows, N=out cols).
// ---------------------------------------------------------------------------
__global__ void mpo_reduce_cores(const float* __restrict__ first,
                                 const float* __restrict__ middle,
                                 const float* __restrict__ last,
                                 float* __restrict__ w) {
  const int t = threadIdx.x;          // 0..255 -> (in, out) pair
  const int i = t >> 4;
  const int k = t & 15;

  float accF = 0.f, accL = 0.f;
  for (int j = 0; j < RANK; ++j) {
    accF += first[(i * RANK + j) * LEG + k];
    accL += last [(i * RANK + j) * LEG + k];
  }
  float accM = 0.f;
  const float* mp = middle + (size_t)i * RANK * RANK * LEG + k;
  for (int rc = 0; rc < RANK * RANK; ++rc) accM += mp[(size_t)rc * LEG];

  w[t]        = accL;
  w[256 + t]  = accM;
  w[512 + t]  = accF;
}

// ---------------------------------------------------------------------------
// Kernel 2: fused triple mode-product, one HBM read + one HBM write.
//   stage : TDM tensor_load_to_lds per wave, hardware row-padding -> RPAD=18
//   step1 : rows (s,a,b) x L  -> rows (s,a,z) col b
//   step2 : rows (s,a,z) x M  -> rows (s,z,y) col a
//   step3 : rows (s,z,y) x F  -> out[s, x, y, z] (+bias)
// Each step: TB*16 WMMA M-tiles, 4 x v_wmma_f32_16x16x4_f32 per tile (K=16).
// ---------------------------------------------------------------------------
__launch_bounds__(256, 1)
__global__ void mpo_apply(const float* __restrict__ x,
                          const float* __restrict__ w,
                          const float* __restrict__ bias,
                          float* __restrict__ out) {
  extern __shared__ float lds[];
  float* bufA = lds;                     // TB * SSTRIDE floats
  float* bufB = lds + TB * SSTRIDE;      // TB * SSTRIDE floats

  const int tid  = threadIdx.x;
  const int lane = tid & 31;
  const int wv   = __builtin_amdgcn_readfirstlane(tid >> 5); // scalar wave id 0..7
  const int h    = lane >> 4;            // lane half (K-range select)
  const int n    = lane & 15;            // N / M-row select

  // ---- B operands for all 3 steps, all 4 K-chunks, in registers ----
  // chunk kc: v0 = W[kc*4 + 2h][n], v1 = W[kc*4 + 2h + 1][n]
  v2f bop[3][4];
#pragma unroll
  for (int s = 0; s < 3; ++s) {
    const float* W = w + s * 256;        // 0:L 1:M 2:F
#pragma unroll
    for (int kc = 0; kc < 4; ++kc) {
      const int r0 = kc * 4 + 2 * h;
      v2f b;
      b.x = W[r0 * 16 + n];
      b.y = W[(r0 + 1) * 16 + n];
      bop[s][kc] = b;
    }
  }

  const size_t base = (size_t)blockIdx.x * TB * 4096;

#if defined(HAVE_TDM)
  // ---- stage via Tensor Data Mover: one 16KB sample per wave --------------
  // D# pad feature inserts 2 DWORDs after every 16 DWORDs -> row stride 18.
  {
    const int s = wv;
    const unsigned long long ga =
        (unsigned long long)(const void*)(x + base + (size_t)s * 4096);
    const unsigned lba =
        (unsigned)(size_t)(const void*)bufA + (unsigned)(s * SSTRIDE * 4);

    u32x4 g0;
    g0.x = 1u;                                        // count=1 valid descriptor
    g0.y = lba;                                       // lds_addr (bytes)
    g0.z = (unsigned)(ga & 0xFFFFFFFFull);            // global_addr[31:0]
    g0.w = (unsigned)((ga >> 32) & 0x01FFFFFFull)     // global_addr[56:32]
         | (2u << 30);                                // type = 2 ("image")

    i32x8 g1;
    g1[0] = (int)((2u << 16)      // data_size = 4 bytes
                | (1u << 20)      // pad_enable
                | (3u << 22)      // pad_interval: 16 DWORDs
                | (1u << 25));    // pad_amount : 2 DWORDs
    g1[1] = (int)(4096u << 16);   // tensor_dim0 = 4096 (lo16)
    g1[2] = (int)(1u << 16);      // tensor_dim0 hi=0 | tensor_dim1 = 1 (lo16)
    g1[3] = (int)(4096u << 16);   // tensor_dim1 hi=0 | tile_dim0 = 4096
    g1[4] = 1;                    // tile_dim1 = 1, tile_dim2 = 0
    g1[5] = 4096;                 // tensor_dim0_stride lo32
    g1[6] = 0;                    // stride0 hi | tensor_dim1_stride lo16
    g1[7] = 0;                    // tensor_dim1_stride hi32

    i32x4 g2 = {0, 0, 0, 0};      // <=2D tensor: groups 2/3 unused
    i32x4 g3 = {0, 0, 0, 0};
#if defined(TDM_ARGS6)
    i32x8 g4 = {0, 0, 0, 0, 0, 0, 0, 0};
    __builtin_amdgcn_tensor_load_to_lds(g0, g1, g2, g3, g4, 0);
#else
    __builtin_amdgcn_tensor_load_to_lds(g0, g1, g2, g3, 0);
#endif
    __builtin_amdgcn_s_wait_tensorcnt(0);
  }
#else
  // ---- fallback: synchronous staged fill ----------------------------------
  for (int idx = tid; idx < TB * 1024; idx += 256) {   // float4 granules
    const int s   = idx >> 10;
    const int e4  = idx & 1023;
    const int row = e4 >> 2;
    const int col = (e4 & 3) * 4;
    v4f v = *(const v4f*)(x + base + (size_t)s * 4096 + row * 16 + col);
    float* p = bufA + s * SSTRIDE + row * RPAD + col;
    p[0] = v.x; p[1] = v.y; p[2] = v.z; p[3] = v.w;
  }
#endif
  __syncthreads();

  // =================== step 1: contract c with L ===================
  for (int t = wv; t < TB * 16; t += 8) {
    const int R = t << 4;
    const float* ap = bufA + (size_t)(R + n) * RPAD + 2 * h;
    v2f a0 = *(const v2f*)(ap + 0);
    v2f a1 = *(const v2f*)(ap + 4);
    v2f a2 = *(const v2f*)(ap + 8);
    v2f a3 = *(const v2f*)(ap + 12);
    v8f c = {0.f, 0.f, 0.f, 0.f, 0.f, 0.f, 0.f, 0.f};
    c = __builtin_amdgcn_wmma_f32_16x16x4_f32(false, a0, false, bop[0][0], (short)0, c, false, false);
    c = __builtin_amdgcn_wmma_f32_16x16x4_f32(false, a1, false, bop[0][1], (short)0, c, false, false);
    c = __builtin_amdgcn_wmma_f32_16x16x4_f32(false, a2, false, bop[0][2], (short)0, c, false, false);
    c = __builtin_amdgcn_wmma_f32_16x16x4_f32(false, a3, false, bop[0][3], (short)0, c, false, false);
#pragma unroll
    for (int v = 0; v < 8; ++v) {        // D(v,lane): row r=R+v+8h (s,a,b), col z=n
      const int r  = R + v + 8 * h;
      const int b  = r & 15;
      const int sa = r >> 4;             // s*16 + a
      bufB[(size_t)(sa * 16 + n) * RPAD + b] = c[v];
    }
  }
  __syncthreads();

  // =================== step 2: contract b with M ===================
  for (int t = wv; t < TB * 16; t += 8) {
    const int R = t << 4;
    const float* ap = bufB + (size_t)(R + n) * RPAD + 2 * h;
    v2f a0 = *(const v2f*)(ap + 0);
    v2f a1 = *(const v2f*)(ap + 4);
    v2f a2 = *(const v2f*)(ap + 8);
    v2f a3 = *(const v2f*)(ap + 12);
    v8f c = {0.f, 0.f, 0.f, 0.f, 0.f, 0.f, 0.f, 0.f};
    c = __builtin_amdgcn_wmma_f32_16x16x4_f32(false, a0, false, bop[1][0], (short)0, c, false, false);
    c = __builtin_amdgcn_wmma_f32_16x16x4_f32(false, a1, false, bop[1][1], (short)0, c, false, false);
    c = __builtin_amdgcn_wmma_f32_16x16x4_f32(false, a2, false, bop[1][2], (short)0, c, false, false);
    c = __builtin_amdgcn_wmma_f32_16x16x4_f32(false, a3, false, bop[1][3], (short)0, c, false, false);
#pragma unroll
    for (int v = 0; v < 8; ++v) {        // row r (s,a,z), col y=n -> dst row (s,z,y) col a
      const int r  = R + v + 8 * h;
      const int z  = r & 15;
      const int sa = r >> 4;
      const int a  = sa & 15;
      const int s  = sa >> 4;
      bufA[(size_t)(((s * 16 + z) * 16) + n) * RPAD + a] = c[v];
    }
  }
  __syncthreads();

  // =================== step 3: contract a with F ===================
  for (int t = wv; t < TB * 16; t += 8) {
    const int R = t << 4;
    const float* ap = bufA + (size_t)(R + n) * RPAD + 2 * h;
    v2f a0 = *(const v2f*)(ap + 0);
    v2f a1 = *(const v2f*)(ap + 4);
    v2f a2 = *(const v2f*)(ap + 8);
    v2f a3 = *(const v2f*)(ap + 12);
    v8f c = {0.f, 0.f, 0.f, 0.f, 0.f, 0.f, 0.f, 0.f};
    c = __builtin_amdgcn_wmma_f32_16x16x4_f32(false, a0, false, bop[2][0], (short)0, c, false, false);
    c = __builtin_amdgcn_wmma_f32_16x16x4_f32(false, a1, false, bop[2][1], (short)0, c, false, false);
    c = __builtin_amdgcn_wmma_f32_16x16x4_f32(false, a2, false, bop[2][2], (short)0, c, false, false);
    c = __builtin_amdgcn_wmma_f32_16x16x4_f32(false, a3, false, bop[2][3], (short)0, c, false, false);
#pragma unroll
    for (int v = 0; v < 8; ++v) {        // row r (s,z,y), col x=n -> out[s][x,y,z]
      const int r = R + v + 8 * h;
      const int y = r & 15;
      const int z = (r >> 4) & 15;
      const int s = r >> 8;
      bufB[(size_t)s * SSTRIDE + (n * 16 + y) * RPAD + z] = c[v];
    }
  }
  __syncthreads();

  // ---- bias add + coalesced store ----
  for (int idx = tid; idx < TB * 1024; idx += 256) {
    const int s = idx >> 10;
    const int o = (idx & 1023) * 4;      // linear offset within 4096
    const float* p = bufB + (size_t)s * SSTRIDE + (o >> 4) * RPAD + (o & 15);
    v4f bv = *(const v4f*)(bias + o);
    v4f r;
    r.x = p[0] + bv.x;
    r.y = p[1] + bv.y;
    r.z = p[2] + bv.z;
    r.w = p[3] + bv.w;
    *(v4f*)(out + base + (size_t)s * 4096 + o) = r;
  }
}

extern "C" void kernel_launch(void* const* d_in, const int* in_sizes, int n_in,
                              void* d_out, int out_size, void* d_ws, size_t ws_size,
                              hipStream_t stream) {
  (void)n_in; (void)out_size; (void)ws_size;
  const float* x      = (const float*)d_in[0];
  const float* first  = (const float*)d_in[1];
  const float* middle = (const float*)d_in[2];
  const float* last   = (const float*)d_in[3];
  const float* bias   = (const float*)d_in[4];
  float* out = (float*)d_out;
  float* w   = (float*)d_ws;             // 768 floats: L | M | F

  hipLaunchKernelGGL(mpo_reduce_cores, dim3(1), dim3(256), 0, stream,
                     first, middle, last, w);

  const int Bn      = in_sizes[0] / 4096;   // batch
  const int nblocks = Bn / TB;              // 16384/8 = 2048
  const size_t shmem = (size_t)2 * TB * SSTRIDE * sizeof(float);  // 288 KB
  hipLaunchKernelGGL(mpo_apply, dim3(nblocks), dim3(256), shmem, stream,
                     x, w, bias, out);
}